// GIN_JK_15925738734176
// MI455X (gfx1250) — compile-verified
//
#include <hip/hip_runtime.h>

#define NND 50000      // N_NODES
#define NED 800000     // N_EDGES
#define DH 128         // D_IN == D_HID
#define NHOPS 4
#define DOUT 40
#define XS 132         // padded LDS row stride (floats): 132%64==4 -> conflict-free A reads

typedef __attribute__((ext_vector_type(2))) float v2f;
typedef __attribute__((ext_vector_type(8))) float v8f;

__device__ __forceinline__ v8f wmma_f32(v2f a, v2f b, v8f c) {
  // D = A(16x4,f32) * B(4x16,f32) + C(16x16,f32)  -> v_wmma_f32_16x16x4_f32
  return __builtin_amdgcn_wmma_f32_16x16x4_f32(false, a, false, b, (short)0, c,
                                               false, false);
}

__global__ __launch_bounds__(256) void zero_k(float* __restrict__ p, int n4) {
  int i = blockIdx.x * 256 + threadIdx.x;
  if (i < n4) ((float4*)p)[i] = make_float4(0.f, 0.f, 0.f, 0.f);
}

// One wave per edge: lane handles 4 consecutive features. agg[dst] += h[src]
__global__ __launch_bounds__(256) void scatter_add_k(
    const float* __restrict__ h, const int* __restrict__ src,
    const int* __restrict__ dst, float* __restrict__ agg, int nE) {
  int e = blockIdx.x * 8 + (threadIdx.x >> 5);
  if (e >= nE) return;
  int lane = threadIdx.x & 31;
  int s = src[e];
  int d = dst[e];
  const float4 v = *(const float4*)(h + (size_t)s * DH + lane * 4);
  float* ap = agg + (size_t)d * DH + lane * 4;
  atomicAdd(ap + 0, v.x);
  atomicAdd(ap + 1, v.y);
  atomicAdd(ap + 2, v.z);
  atomicAdd(ap + 3, v.w);
}

// Per block: 16 rows. 8 waves, wave w computes output cols [16w, 16w+16).
// hout = relu( relu( ((1+eps)h + agg) @ W1 + b1 ) @ W2 + b2 )
__global__ __launch_bounds__(256) void gin_mlp_k(
    const float* __restrict__ hprev, const float* __restrict__ agg,
    const float* __restrict__ eps, int hop,
    const float* __restrict__ W1, const float* __restrict__ b1,
    const float* __restrict__ W2, const float* __restrict__ b2,
    float* __restrict__ hout) {
  __shared__ float Xs[16 * XS];
  __shared__ float Zs[16 * XS];
  const int tid = threadIdx.x;
  const int wave = tid >> 5;
  const int lane = tid & 31;
  const int l16 = lane & 15;          // A row / B,C column within tile
  const int koff = (lane >> 4) << 1;  // 0 or 2 : K-pair for A/B fragments
  const int mAdd = (lane >> 4) << 3;  // 0 or 8 : C row offset
  const int colBase = wave << 4;
  const size_t rowBase = (size_t)blockIdx.x * 16;
  const float e1 = 1.0f + eps[hop];

  // Stage X = (1+eps)*h + agg : 16 x 128 tile into LDS
  for (int i = tid; i < 16 * DH / 4; i += 256) {
    int r = i >> 5;
    int c = (i & 31) << 2;
    size_t g = (rowBase + r) * DH + c;
    float4 hv = *(const float4*)(hprev + g);
    float4 av = *(const float4*)(agg + g);
    float* xp = &Xs[r * XS + c];
    xp[0] = e1 * hv.x + av.x;
    xp[1] = e1 * hv.y + av.y;
    xp[2] = e1 * hv.z + av.z;
    xp[3] = e1 * hv.w + av.w;
  }
  __syncthreads();

  // GEMM1: Z = relu(X @ W1 + b1)
  v8f acc = {};
  for (int k = 0; k < DH; k += 4) {
    v2f a = *(const v2f*)&Xs[l16 * XS + k + koff];
    v2f b;
    b.x = W1[(k + koff) * DH + colBase + l16];
    b.y = W1[(k + koff + 1) * DH + colBase + l16];
    acc = wmma_f32(a, b, acc);
  }
  {
    float bias = b1[colBase + l16];
#pragma unroll
    for (int v = 0; v < 8; v++) {
      float val = acc[v] + bias;
      Zs[(v + mAdd) * XS + colBase + l16] = val > 0.f ? val : 0.f;
    }
  }
  __syncthreads();

  // GEMM2: hout = relu(Z @ W2 + b2)
  v8f acc2 = {};
  for (int k = 0; k < DH; k += 4) {
    v2f a = *(const v2f*)&Zs[l16 * XS + k + koff];
    v2f b;
    b.x = W2[(k + koff) * DH + colBase + l16];
    b.y = W2[(k + koff + 1) * DH + colBase + l16];
    acc2 = wmma_f32(a, b, acc2);
  }
  {
    float bias = b2[colBase + l16];
#pragma unroll
    for (int v = 0; v < 8; v++) {
      float val = acc2[v] + bias;
      hout[(rowBase + v + mAdd) * DH + colBase + l16] = val > 0.f ? val : 0.f;
    }
  }
}

// out = relu(cat @ Wf1 + bf1) @ Wf2 + bf2 ; cat = [feat | H0 | H1 | H2 | H3]
__global__ __launch_bounds__(256) void final_k(
    const float* __restrict__ feat, const float* __restrict__ H,
    const float* __restrict__ Wf1, const float* __restrict__ bf1,
    const float* __restrict__ Wf2, const float* __restrict__ bf2,
    float* __restrict__ out) {
  __shared__ float Xs[16 * XS];
  __shared__ float Ms[16 * XS];
  const int tid = threadIdx.x;
  const int wave = tid >> 5;
  const int lane = tid & 31;
  const int l16 = lane & 15;
  const int koff = (lane >> 4) << 1;
  const int mAdd = (lane >> 4) << 3;
  const int colBase = wave << 4;
  const size_t rowBase = (size_t)blockIdx.x * 16;

  // GEMM over 5 K-blocks of 128
  v8f acc = {};
  for (int kb = 0; kb < 1 + NHOPS; kb++) {
    const float* srcp =
        (kb == 0) ? feat : (H + (size_t)(kb - 1) * NND * DH);
    for (int i = tid; i < 16 * DH / 4; i += 256) {
      int r = i >> 5;
      int c = (i & 31) << 2;
      *(float4*)&Xs[r * XS + c] = *(const float4*)(srcp + (rowBase + r) * DH + c);
    }
    __syncthreads();
    for (int k = 0; k < DH; k += 4) {
      v2f a = *(const v2f*)&Xs[l16 * XS + k + koff];
      int kg = kb * DH + k + koff;
      v2f b;
      b.x = Wf1[kg * DH + colBase + l16];
      b.y = Wf1[(kg + 1) * DH + colBase + l16];
      acc = wmma_f32(a, b, acc);
    }
    __syncthreads();
  }
  {
    float bias = bf1[colBase + l16];
#pragma unroll
    for (int v = 0; v < 8; v++) {
      float val = acc[v] + bias;
      Ms[(v + mAdd) * XS + colBase + l16] = val > 0.f ? val : 0.f;
    }
  }
  __syncthreads();

  // Second GEMM: 40 output cols -> 3 col-tiles (waves 0..2), masked to 40.
  if (wave < 3) {
    const int cB = wave << 4;
    const int col = cB + l16;
    const int cc = col < DOUT ? col : 0;
    const float msk = col < DOUT ? 1.f : 0.f;  // value-select: EXEC stays full
    v8f acc2 = {};
    for (int k = 0; k < DH; k += 4) {
      v2f a = *(const v2f*)&Ms[l16 * XS + k + koff];
      v2f b;
      b.x = Wf2[(k + koff) * DOUT + cc] * msk;
      b.y = Wf2[(k + koff + 1) * DOUT + cc] * msk;
      acc2 = wmma_f32(a, b, acc2);
    }
    if (col < DOUT) {
      float bias = bf2[col];
#pragma unroll
      for (int v = 0; v < 8; v++) {
        out[(rowBase + v + mAdd) * DOUT + col] = acc2[v] + bias;
      }
    }
  }
}

extern "C" void kernel_launch(void* const* d_in, const int* in_sizes, int n_in,
                              void* d_out, int out_size, void* d_ws,
                              size_t ws_size, hipStream_t stream) {
  const float* feat = (const float*)d_in[0];
  const int* src = (const int*)d_in[1];
  const int* dst = (const int*)d_in[2];
  const float* eps = (const float*)d_in[3];
  const float* W1 = (const float*)d_in[4];
  const float* b1 = (const float*)d_in[5];
  const float* W2 = (const float*)d_in[6];
  const float* b2 = (const float*)d_in[7];
  const float* Wf1 = (const float*)d_in[8];
  const float* bf1 = (const float*)d_in[9];
  const float* Wf2 = (const float*)d_in[10];
  const float* bf2 = (const float*)d_in[11];
  float* out = (float*)d_out;

  // Workspace: agg [N,128] then H [4,N,128]  (total 128 MB)
  float* agg = (float*)d_ws;
  float* H = agg + (size_t)NND * DH;

  const int rowBlocks = NND / 16;          // 3125 (exact)
  const int edgeBlocks = (NED + 7) / 8;    // one wave per edge, 8 waves/block
  const int zeroN4 = NND * DH / 4;
  const int zeroBlocks = (zeroN4 + 255) / 256;

  const float* prev = feat;
  for (int i = 0; i < NHOPS; i++) {
    zero_k<<<zeroBlocks, 256, 0, stream>>>(agg, zeroN4);
    scatter_add_k<<<edgeBlocks, 256, 0, stream>>>(prev, src, dst, agg, NED);
    float* hout = H + (size_t)i * NND * DH;
    gin_mlp_k<<<rowBlocks, 256, 0, stream>>>(
        prev, agg, eps, i, W1 + (size_t)i * DH * DH, b1 + (size_t)i * DH,
        W2 + (size_t)i * DH * DH, b2 + (size_t)i * DH, hout);
    prev = hout;
  }
  final_k<<<rowBlocks, 256, 0, stream>>>(feat, H, Wf1, bf1, Wf2, bf2, out);
}